// KerasCustomMappingLayer_70317204570634
// MI455X (gfx1250) — compile-verified
//
#include <hip/hip_runtime.h>
#include <math.h>

// ---------------------------------------------------------------------------
// KerasCustomMappingLayer "construct_map" for MI455X (gfx1250).
// Memory-bound streaming scan: 160 B in + 240 B out per batch element
// (~420 MB total -> ~18 us at 23.3 TB/s).
// Staging: linear (pitch-40) LDS tile so async global->LDS b128 loads use a
// single base + literal immediate offsets (IOFFSET applies to both addresses).
// Compute: fully unrolled 10-step recurrence, hw sqrt + single reciprocal per
// normalize pair, non-temporal vector stores of 60 contiguous outputs.
// ---------------------------------------------------------------------------

#define NGS 10
#define ROWLEN (4 * NGS)        // 40 floats per batch row
#define TPB 256                 // 8 wave32 per block

typedef __attribute__((ext_vector_type(2))) float v2f;
typedef __attribute__((ext_vector_type(4))) int   v4i;

#if defined(__HIP_DEVICE_COMPILE__)
#if __has_builtin(__builtin_amdgcn_global_load_async_to_lds_b128)
#define HAVE_ASYNC_B128 1
#else
#define HAVE_ASYNC_B128 0
#endif
#if __has_builtin(__builtin_amdgcn_global_load_async_to_lds_b32)
#define HAVE_ASYNC_B32 1
#else
#define HAVE_ASYNC_B32 0
#endif
#if __has_builtin(__builtin_amdgcn_sqrtf)
#define SQRTF(x) __builtin_amdgcn_sqrtf(x)
#else
#define SQRTF(x) sqrtf(x)
#endif
#else
#define HAVE_ASYNC_B128 0
#define HAVE_ASYNC_B32 0
#define SQRTF(x) sqrtf(x)
#endif

#define INV384 (1.0f / 384.0f)   // compile-time correctly rounded

__global__ __launch_bounds__(TPB) void osu_construct_map(
    const float* __restrict__ var,         // (B, 40)
    const float* __restrict__ slider_len,  // (10)
    const float* __restrict__ slider_cos,  // (10)
    const float* __restrict__ slider_sin,  // (10)
    const float* __restrict__ note_dist,   // (10)
    const float* __restrict__ tick_diff,   // (10)
    const float* __restrict__ start_pos,   // (2)
    const int*   __restrict__ is_slider,   // (10)
    float* __restrict__ out,               // (B, 10, 6)
    int B)
{
    __shared__ float tile[TPB * ROWLEN];   // linear: LDS offset == global offset

    const int tid = threadIdx.x;
    const long long rowBase = (long long)blockIdx.x * TPB;
    const long long gBase   = rowBase * ROWLEN;

    if (rowBase + TPB <= (long long)B) {
        // ---- Full tile: guard-free staging, one base + literal offsets ----
#if HAVE_ASYNC_B128
        {
            const float* gsrc = var + gBase + tid * 4;   // lane base (16B chunk)
            float* lds = &tile[tid * 4];
            // 10 x b128; IOFFSET = i*4096 applies to global AND LDS address
#define STAGE128(OFF)                                                      \
            __builtin_amdgcn_global_load_async_to_lds_b128(                \
                (__attribute__((address_space(1))) v4i*)gsrc,              \
                (__attribute__((address_space(3))) v4i*)lds, (OFF), 0)
            STAGE128(0 * 4096); STAGE128(1 * 4096); STAGE128(2 * 4096);
            STAGE128(3 * 4096); STAGE128(4 * 4096); STAGE128(5 * 4096);
            STAGE128(6 * 4096); STAGE128(7 * 4096); STAGE128(8 * 4096);
            STAGE128(9 * 4096);
#undef STAGE128
        }
#elif HAVE_ASYNC_B32
        {
            const float* gsrc = var + gBase + tid;
            float* lds = &tile[tid];
            // 40 x b32; IOFFSET = i*1024
#define STAGE32(OFF)                                                       \
            __builtin_amdgcn_global_load_async_to_lds_b32(                 \
                (__attribute__((address_space(1))) int*)gsrc,              \
                (__attribute__((address_space(3))) int*)lds, (OFF), 0)
            STAGE32(0 * 1024);  STAGE32(1 * 1024);  STAGE32(2 * 1024);
            STAGE32(3 * 1024);  STAGE32(4 * 1024);  STAGE32(5 * 1024);
            STAGE32(6 * 1024);  STAGE32(7 * 1024);  STAGE32(8 * 1024);
            STAGE32(9 * 1024);  STAGE32(10 * 1024); STAGE32(11 * 1024);
            STAGE32(12 * 1024); STAGE32(13 * 1024); STAGE32(14 * 1024);
            STAGE32(15 * 1024); STAGE32(16 * 1024); STAGE32(17 * 1024);
            STAGE32(18 * 1024); STAGE32(19 * 1024); STAGE32(20 * 1024);
            STAGE32(21 * 1024); STAGE32(22 * 1024); STAGE32(23 * 1024);
            STAGE32(24 * 1024); STAGE32(25 * 1024); STAGE32(26 * 1024);
            STAGE32(27 * 1024); STAGE32(28 * 1024); STAGE32(29 * 1024);
            STAGE32(30 * 1024); STAGE32(31 * 1024); STAGE32(32 * 1024);
            STAGE32(33 * 1024); STAGE32(34 * 1024); STAGE32(35 * 1024);
            STAGE32(36 * 1024); STAGE32(37 * 1024); STAGE32(38 * 1024);
            STAGE32(39 * 1024);
#undef STAGE32
        }
#else
#pragma unroll
        for (int i = 0; i < ROWLEN; ++i) {
            const int j = i * TPB + tid;
            tile[j] = var[gBase + j];
        }
#endif
#if HAVE_ASYNC_B128 || HAVE_ASYNC_B32
#if __has_builtin(__builtin_amdgcn_s_wait_asynccnt)
        __builtin_amdgcn_s_wait_asynccnt(0);
#else
        asm volatile("s_wait_asynccnt 0" ::: "memory");
#endif
#endif
    } else {
        // ---- Tail tile (at most one block): compact guarded loop ----
        const long long gTotal = (long long)B * ROWLEN;
#pragma unroll 1
        for (int i = 0; i < ROWLEN; ++i) {
            const int j = i * TPB + tid;
            const long long g = gBase + j;
            tile[j] = (g < gTotal) ? var[g] : 0.0f;
        }
    }
    __syncthreads();

    const long long b = rowBase + tid;
    if (b >= B) return;

    const float* __restrict__ r = &tile[tid * ROWLEN];
    float px = start_pos[0];
    float py = start_pos[1];
    float* __restrict__ o = out + b * (long long)(NGS * 6);

#pragma unroll
    for (int k = 0; k < NGS; ++k) {
        // Per-element inputs for this step (2x ds_load_2addr)
        const float c0 = r[k];            // var[:, k]        (cos_raw low / vk)
        const float c1 = r[NGS + k];      // var[:, 10+k]     (cos_raw high)
        const float s0 = r[2 * NGS + k];  // var[:, 20+k]     (sin_raw low / vk2)
        const float s1 = r[3 * NGS + k];  // var[:, 30+k]     (sin_raw high)

        // normalize: hw sqrt (<=1ulp) + one IEEE reciprocal per pair
        const float inv0 = 1.0f / SQRTF(c0 * c0 + s0 * s0);
        const float inv1 = 1.0f / SQRTF(c1 * c1 + s1 * s1);
        const float ck  = c0 * inv0;
        const float sk  = s0 * inv0;
        const float ckh = c1 * inv1;
        const float skh = s1 * inv1;

        // Wave-uniform per-step scalars (scalar-cached loads)
        const float l  = 1.0f * note_dist[k];       // LMUL = 1
        const float wl = 25.6f  + 0.5f * l;         // 0.05*512 + l/2
        const float wr = 486.4f - 0.5f * l;         // 0.95*512 - l/2
        const float wt = 19.2f  + 0.5f * l;         // 0.05*384 + l/2
        const float wb = 364.8f - 0.5f * l;         // 0.95*384 - l/2
        const float rr = (tick_diff[k] <= 1.0f) ? 0.0f : 1.0f;  // rerand
        const float nr = 1.0f - rr;

        const float rerand_x = 256.0f + 256.0f * c0;
        const float rerand_y = 192.0f + 192.0f * s0;
        const float dx = l * ck, dy = l * sk;
        const float adx = fabsf(dx), ady = fabsf(dy);

        // Wall masks, arithmetic form of the reference
        const float wv_l  = (px < wl) ? 1.0f : 0.0f;
        const float wv_r  = (px > wr) ? 1.0f : 0.0f;
        const float wv_xm = ((px > wl) ? 1.0f : 0.0f) * ((px < wr) ? 1.0f : 0.0f);
        const float wv_t  = (py < wt) ? 1.0f : 0.0f;
        const float wv_b  = (py > wb) ? 1.0f : 0.0f;
        const float wv_ym = ((py > wt) ? 1.0f : 0.0f) * ((py < wb) ? 1.0f : 0.0f);

        const float x_delta = adx * wv_l - adx * wv_r + dx * wv_xm;
        const float y_delta = ady * wv_t - ady * wv_b + dy * wv_ym;

        const float _x = rr * rerand_x + nr * (px + x_delta);
        const float _y = rr * rerand_y + nr * (py + y_delta);

        const float sc = slider_cos[k];
        const float ss = slider_sin[k];
        const float L  = slider_len[k];
        const float _oa = ckh * sc - skh * ss;
        const float _ob = ckh * ss + skh * sc;

        const float a2 = rr * ckh + nr * ck;
        const float b2 = rr * skh + nr * sk;

        const bool slider = (is_slider[k] != 0);

        const float o0 = _x / 512.0f;                 // exact (power of two)
        const float o1 = _y * INV384;
        const float o2 = slider ? _oa : a2;
        const float o3 = slider ? _ob : b2;
        const float o4 = slider ? ((_x + ckh * L) / 512.0f) : o0;
        const float o5 = slider ? ((_y + skh * L) * INV384) : o1;

        // 60 contiguous floats per element; write-once -> non-temporal stores
        v2f* op = reinterpret_cast<v2f*>(o + k * 6);
        const v2f p01 = {o0, o1};
        const v2f p23 = {o2, o3};
        const v2f p45 = {o4, o5};
        __builtin_nontemporal_store(p01, op + 0);
        __builtin_nontemporal_store(p23, op + 1);
        __builtin_nontemporal_store(p45, op + 2);

        // NFSE == False  =>  carried position is (_x, _y) for both branches
        px = _x;
        py = _y;
    }
}

extern "C" void kernel_launch(void* const* d_in, const int* in_sizes, int n_in,
                              void* d_out, int out_size, void* d_ws, size_t ws_size,
                              hipStream_t stream) {
    const float* var        = (const float*)d_in[0];
    const float* slider_len = (const float*)d_in[1];
    const float* slider_cos = (const float*)d_in[2];
    const float* slider_sin = (const float*)d_in[3];
    const float* note_dist  = (const float*)d_in[4];
    const float* tickd      = (const float*)d_in[5];
    const float* start_pos  = (const float*)d_in[6];
    const int*   is_slider  = (const int*)d_in[7];
    float* out = (float*)d_out;

    const int B = in_sizes[0] / ROWLEN;          // 1048576 for the given harness
    const int grid = (B + TPB - 1) / TPB;

    osu_construct_map<<<grid, TPB, 0, stream>>>(var, slider_len, slider_cos,
                                                slider_sin, note_dist, tickd,
                                                start_pos, is_slider, out, B);
    (void)n_in; (void)out_size; (void)d_ws; (void)ws_size;
}